// BasicLayer_33870112096814
// MI455X (gfx1250) — compile-verified
//
#include <hip/hip_runtime.h>
#include <hip/hip_bf16.h>
#include <math.h>

#define NPTS   8192
#define MBALL  64
#define NBALL  128      // NPTS / MBALL
#define DIM    64
#define NHEAD  8
#define EH     8        // DIM / NHEAD
#define HID    256      // DIM * MLP
#define EPSRMS 1.1920929e-07f

typedef __attribute__((ext_vector_type(2))) float v2f;
typedef __attribute__((ext_vector_type(8))) float v8f;

// ---------------------------------------------------------------------------
// Kernel 1: per-ball prep.  One block per ball, 64 threads.
//   x1[n]  = rmsnorm(x[n])*norm1_w + (pos[n] - ball_pos_mean)
//   keys[b]= mean over ball of x1  (routing keys; head slices of this)
// ---------------------------------------------------------------------------
__global__ void prep_kernel(const float* __restrict__ x,
                            const float* __restrict__ pos,
                            const float* __restrict__ norm1w,
                            float* __restrict__ x1,
                            float* __restrict__ keys) {
    __shared__ float s_pm[DIM];
    __shared__ float s_x1[MBALL * DIM];
    const int ball = blockIdx.x;
    const int tid  = threadIdx.x;          // 0..63
    const int base = ball * MBALL;         // first point of ball

    // pos mean over the ball: thread d handles dim d
    {
        float s = 0.f;
        #pragma unroll 4
        for (int m = 0; m < MBALL; ++m)
            s += pos[(base + m) * DIM + tid];
        s_pm[tid] = s * (1.0f / MBALL);
    }
    __syncthreads();

    // thread m handles point n = base + m
    {
        const int n = base + tid;
        float ss = 0.f;
        float xr[DIM];
        #pragma unroll 8
        for (int d = 0; d < DIM; ++d) {
            float v = x[n * DIM + d];
            xr[d] = v;
            ss += v * v;
        }
        const float r = rsqrtf(ss * (1.0f / DIM) + EPSRMS);
        #pragma unroll 8
        for (int d = 0; d < DIM; ++d) {
            float v = xr[d] * r * norm1w[d] + pos[n * DIM + d] - s_pm[d];
            x1[n * DIM + d] = v;
            s_x1[tid * DIM + d] = v;
        }
    }
    __syncthreads();

    // ball mean of x1: thread d handles dim d
    {
        float s = 0.f;
        #pragma unroll 4
        for (int m = 0; m < MBALL; ++m)
            s += s_x1[m * DIM + tid];
        keys[ball * DIM + tid] = s * (1.0f / MBALL);
    }
}

// ---------------------------------------------------------------------------
// Kernel 2: routing.  One thread per (h, n).  Top-2 argmax of q . keys
// (softmax is monotonic; forward gates are exactly 1.0 -> values unneeded).
// ---------------------------------------------------------------------------
__global__ void route_kernel(const float* __restrict__ x1,
                             const float* __restrict__ keys,
                             int* __restrict__ idx) {
    const int gid = blockIdx.x * blockDim.x + threadIdx.x;   // 0..H*N-1
    const int h = gid >> 13;          // / NPTS
    const int n = gid & (NPTS - 1);
    float q[EH];
    #pragma unroll
    for (int e = 0; e < EH; ++e) q[e] = x1[n * DIM + h * EH + e];

    float best1 = -INFINITY, best2 = -INFINITY;
    int i1 = 0, i2 = 0;
    for (int b = 0; b < NBALL; ++b) {
        const float* kp = keys + b * DIM + h * EH;
        float s = 0.f;
        #pragma unroll
        for (int e = 0; e < EH; ++e) s += q[e] * kp[e];
        if (s > best1) { best2 = best1; i2 = i1; best1 = s; i1 = b; }
        else if (s > best2) { best2 = s; i2 = b; }
    }
    idx[gid * 2 + 0] = i1;
    idx[gid * 2 + 1] = i2;
}

// ---------------------------------------------------------------------------
// Kernel 3: gathered attention.  One wave32 per (h, n); 128 K/V rows split
// 4 per lane.  hmid = x + attn_out (head-major matches x layout).
// ---------------------------------------------------------------------------
__global__ void attend_kernel(const float* __restrict__ x1,
                              const int* __restrict__ idx,
                              const float* __restrict__ x,
                              float* __restrict__ hmid) {
    const int wave = blockIdx.x * (blockDim.x >> 5) + (threadIdx.x >> 5);
    const int lane = threadIdx.x & 31;
    const int h = wave >> 13;
    const int n = wave & (NPTS - 1);

    float q[EH];
    #pragma unroll
    for (int e = 0; e < EH; ++e) q[e] = x1[n * DIM + h * EH + e];

    const int b0 = idx[wave * 2 + 0];
    const int b1 = idx[wave * 2 + 1];

    float kk[4][EH];
    float sc[4];
    #pragma unroll
    for (int c = 0; c < 4; ++c) {
        const int j = lane + 32 * c;          // 0..127
        const int ball = (j < 64) ? b0 : b1;
        const int p = ball * MBALL + (j & 63);
        float s = 0.f;
        #pragma unroll
        for (int e = 0; e < EH; ++e) {
            float kv = x1[p * DIM + h * EH + e];
            kk[c][e] = kv;
            s += q[e] * kv;
        }
        sc[c] = s * 0.35355339059327373f;     // EH^-0.5
    }

    float mx = fmaxf(fmaxf(sc[0], sc[1]), fmaxf(sc[2], sc[3]));
    #pragma unroll
    for (int off = 16; off > 0; off >>= 1)
        mx = fmaxf(mx, __shfl_xor(mx, off, 32));

    float den = 0.f;
    float w[4];
    #pragma unroll
    for (int c = 0; c < 4; ++c) { w[c] = __expf(sc[c] - mx); den += w[c]; }
    #pragma unroll
    for (int off = 16; off > 0; off >>= 1)
        den += __shfl_xor(den, off, 32);

    float o[EH];
    #pragma unroll
    for (int e = 0; e < EH; ++e) {
        float a = w[0]*kk[0][e] + w[1]*kk[1][e] + w[2]*kk[2][e] + w[3]*kk[3][e];
        #pragma unroll
        for (int off = 16; off > 0; off >>= 1)
            a += __shfl_xor(a, off, 32);
        o[e] = a;
    }

    if (lane == 0) {
        const float inv = __builtin_amdgcn_rcpf(den);   // v_rcp_f32, ~1 ulp
        #pragma unroll
        for (int e = 0; e < EH; ++e)
            hmid[n * DIM + h * EH + e] = x[n * DIM + h * EH + e] + o[e] * inv;
    }
}

// ---------------------------------------------------------------------------
// Kernel 4: second RMSNorm.  One thread per point.
// ---------------------------------------------------------------------------
__global__ void rms2_kernel(const float* __restrict__ hmid,
                            const float* __restrict__ norm2w,
                            float* __restrict__ x2) {
    const int n = blockIdx.x * blockDim.x + threadIdx.x;
    float buf[DIM];
    float ss = 0.f;
    #pragma unroll 8
    for (int d = 0; d < DIM; ++d) {
        float v = hmid[n * DIM + d];
        buf[d] = v;
        ss += v * v;
    }
    const float r = rsqrtf(ss * (1.0f / DIM) + EPSRMS);
    #pragma unroll 8
    for (int d = 0; d < DIM; ++d)
        x2[n * DIM + d] = buf[d] * r * norm2w[d];
}

// ---------------------------------------------------------------------------
// Kernel 5: SwiGLU MLP with V_WMMA_F32_16X16X4_F32.
// 256 threads = 8 wave32 per block; each block does a 16-row tile.
// Fragment layouts per CDNA5 ISA 7.12.2 (32-bit A 16x4 / B 4x16 / C-D 16x16).
// ---------------------------------------------------------------------------
__global__ void mlp_kernel(const float* __restrict__ x2,
                           const float* __restrict__ hmid,
                           const float* __restrict__ w1w, const float* __restrict__ w1b,
                           const float* __restrict__ w2w, const float* __restrict__ w2b,
                           const float* __restrict__ w3w, const float* __restrict__ w3b,
                           float* __restrict__ out) {
    __shared__ float sA[16 * DIM];        // activation tile (16 x 64)
    __shared__ float sG[16 * (HID + 4)];  // gated hidden tile (16 x 256, padded)
    const int GSTR = HID + 4;

    const int row0 = blockIdx.x * 16;
    const int tid  = threadIdx.x;
    const int wv   = tid >> 5;            // wave id 0..7 (wave-uniform)
    const int lane = tid & 31;

    // stage activation tile (coalesced: 4 floats per thread)
    #pragma unroll
    for (int k = 0; k < 4; ++k) {
        int i = tid + k * 256;            // 0..1023 over 16x64
        sA[i] = x2[row0 * DIM + i];
    }
    __syncthreads();

    const int mrow  = lane & 15;          // A-matrix M index
    const int khalf = (lane >> 4) << 1;   // K sub-offset: 0 (lanes 0-15) / 2 (16-31)
    const int ncol  = lane & 15;          // B/C/D N index within tile
    const int rbase = (lane < 16) ? 0 : 8;

    // ----- up projections + SwiGLU: each wave owns 32 hidden columns -----
    #pragma unroll
    for (int t = 0; t < 2; ++t) {
        const int cb  = wv * 32 + t * 16;      // hidden-column tile base
        const int col = cb + ncol;
        v8f c1 = {}; v8f c2 = {};
        #pragma unroll 4
        for (int k = 0; k < DIM; k += 4) {
            v2f a;
            a.x = sA[mrow * DIM + k + khalf];
            a.y = sA[mrow * DIM + k + khalf + 1];
            const float* p1 = w1w + col * DIM + k + khalf;   // B[kk][o] = w[o][kk]
            const float* p2 = w2w + col * DIM + k + khalf;
            v2f b1; b1.x = p1[0]; b1.y = p1[1];
            v2f b2; b2.x = p2[0]; b2.y = p2[1];
            c1 = __builtin_amdgcn_wmma_f32_16x16x4_f32(false, a, false, b1,
                                                       (short)0, c1, false, false);
            c2 = __builtin_amdgcn_wmma_f32_16x16x4_f32(false, a, false, b2,
                                                       (short)0, c2, false, false);
        }
        const float bb1 = w1b[col];
        const float bb2 = w2b[col];
        #pragma unroll
        for (int r = 0; r < 8; ++r) {
            float aa = c1[r] + bb1;                       // gate path (w1)
            float vv = c2[r] + bb2;                       // value path (w2)
            // silu(aa) = aa * sigmoid(aa); v_rcp_f32 instead of IEEE divide
            float si = aa * __builtin_amdgcn_rcpf(1.0f + __expf(-aa));
            sG[(rbase + r) * GSTR + col] = vv * si;
        }
    }
    __syncthreads();

    // ----- down projection: waves 0-3 own 16 output columns each -----
    if (wv < 4) {                                         // wave-uniform branch
        const int col = wv * 16 + ncol;
        v8f c3 = {};
        #pragma unroll 4
        for (int k = 0; k < HID; k += 4) {
            v2f a;
            a.x = sG[mrow * GSTR + k + khalf];
            a.y = sG[mrow * GSTR + k + khalf + 1];
            const float* p3 = w3w + col * HID + k + khalf;
            v2f b; b.x = p3[0]; b.y = p3[1];
            c3 = __builtin_amdgcn_wmma_f32_16x16x4_f32(false, a, false, b,
                                                       (short)0, c3, false, false);
        }
        const float bb = w3b[col];
        #pragma unroll
        for (int r = 0; r < 8; ++r) {
            const int row = row0 + rbase + r;
            out[row * DIM + col] = c3[r] + bb + hmid[row * DIM + col];
        }
    }
}

// ---------------------------------------------------------------------------
extern "C" void kernel_launch(void* const* d_in, const int* in_sizes, int n_in,
                              void* d_out, int out_size, void* d_ws, size_t ws_size,
                              hipStream_t stream) {
    const float* x      = (const float*)d_in[0];
    const float* pos    = (const float*)d_in[1];
    const float* norm1w = (const float*)d_in[2];
    const float* norm2w = (const float*)d_in[3];
    const float* w1w    = (const float*)d_in[4];
    const float* w1b    = (const float*)d_in[5];
    const float* w2w    = (const float*)d_in[6];
    const float* w2b    = (const float*)d_in[7];
    const float* w3w    = (const float*)d_in[8];
    const float* w3b    = (const float*)d_in[9];
    float* out = (float*)d_out;

    float* ws   = (float*)d_ws;
    float* x1   = ws;                        // N*DIM
    float* keys = x1 + NPTS * DIM;           // NBALL*DIM
    float* hmid = keys + NBALL * DIM;        // N*DIM
    float* x2   = hmid + NPTS * DIM;         // N*DIM
    int*   idx  = (int*)(x2 + NPTS * DIM);   // H*N*2 ints

    prep_kernel<<<NBALL, 64, 0, stream>>>(x, pos, norm1w, x1, keys);
    route_kernel<<<(NHEAD * NPTS) / 256, 256, 0, stream>>>(x1, keys, idx);
    attend_kernel<<<(NHEAD * NPTS) / 8, 256, 0, stream>>>(x1, idx, x, hmid);
    rms2_kernel<<<NPTS / 256, 256, 0, stream>>>(hmid, norm2w, x2);
    mlp_kernel<<<NPTS / 16, 256, 0, stream>>>(x2, hmid, w1w, w1b, w2w, w2b,
                                              w3w, w3b, out);
}